// ParamRecoveryModelSLL_65575560675420
// MI455X (gfx1250) — compile-verified
//
#include <hip/hip_runtime.h>
#include <math.h>

#if defined(__HIP_DEVICE_COMPILE__) && \
    __has_builtin(__builtin_amdgcn_global_load_async_to_lds_b128) && \
    __has_builtin(__builtin_amdgcn_s_wait_asynccnt)
#define USE_ASYNC_LDS 1
#else
#define USE_ASYNC_LDS 0
#endif

namespace {

constexpr float FS_HZ = 44100.0f;

// RC time-constant bounds (seconds), from the reference header.
constexpr float T_AF_MIN = (float)(820.0    * 4.7e-07 * 0.8);
constexpr float T_AF_MAX = (float)(270000.0 * 4.7e-07 * 1.2);
constexpr float T_AS_MIN = (float)(820.0    * 6.8e-06 * 0.8);
constexpr float T_AS_MAX = (float)(270000.0 * 6.8e-06 * 100.0);
constexpr float T_SF_MIN = (float)(91000.0  * 4.7e-07 * 0.8);
constexpr float T_SF_MAX = (float)(1200000.0* 4.7e-07 * 1.2);
constexpr float T_SS_MIN = (float)(750000.0 * 6.8e-06 * 0.8);
constexpr float T_SS_MAX = (float)(750000.0 * 6.8e-06 * 100.0);

__device__ __forceinline__ float pole_coeff(float uu, float tmin, float tmax) {
    float sig  = 1.0f / (1.0f + expf(-uu));          // sigmoid
    float secs = tmin + (tmax - tmin) * sig;
    return expf(-1.0f / (FS_HZ * secs));
}

struct ScanState { float u, v, e; };  // u=ef/2, v=es/2, e=u+v (=0.5*(ef+es))

struct Params {
    float thresh, slope, fbs;
    float a_af, a_as, a_sf, a_ss;
};

// One compressor sample. Critical dependent chain (5 VALU):
//   fma(t0) -> cmp -> cndmask -> fma(u') -> add(e')
// max / 0.5* / (u-ht) execute in the cmp/cndmask latency shadow.
__device__ __forceinline__ float step(ScanState& st, float s, const Params& P) {
    float t0  = fmaf(-P.fbs, st.e, s);      // (lvl - thresh)*slope, pre-hinge
    bool  att = t0 > st.e;                  // == (max(t0,0) > e) since e >= 0
    float ht  = 0.5f * fmaxf(t0, 0.0f);     // 0.5 * tgt
    float au  = att ? P.a_af : P.a_sf;
    float av  = att ? P.a_as : P.a_ss;
    st.u = fmaf(au, st.u - ht, ht);         // 0.5*ef' = af*(u-ht)+ht
    st.v = fmaf(av, st.v - ht, ht);         // 0.5*es'
    st.e = st.u + st.v;
    return -st.e;                           // y = -0.5*(ef'+es')  (off critical path)
}

__device__ __forceinline__ float4 step4(ScanState& st, const float4& c, const Params& P) {
    float4 o;
    o.x = step(st, (c.x - P.thresh) * P.slope, P);
    o.y = step(st, (c.y - P.thresh) * P.slope, P);
    o.z = step(st, (c.z - P.thresh) * P.slope, P);
    o.w = step(st, (c.w - P.thresh) * P.slope, P);
    return o;
}

__device__ __forceinline__ Params make_params(
    const float* p_thresh, const float* p_ratio, const float* p_fb,
    const float* p_uaf, const float* p_uas, const float* p_usf, const float* p_uss) {
    Params P;
    P.thresh = p_thresh[0];
    float cr = fmaxf(expf(p_ratio[0]) + 1.0f, 1.0f + 1e-4f);
    P.slope = 1.0f - 1.0f / cr;
    float fb = 1.0f / (1.0f + expf(-p_fb[0]));
    fb = fminf(fmaxf(fb, 0.0f), 1.0f);
    P.a_af = pole_coeff(p_uaf[0], T_AF_MIN, T_AF_MAX);
    P.a_as = pole_coeff(p_uas[0], T_AS_MIN, T_AS_MAX);
    P.a_sf = pole_coeff(p_usf[0], T_SF_MIN, T_SF_MAX);
    P.a_ss = pole_coeff(p_uss[0], T_SS_MIN, T_SS_MAX);
    P.fbs  = fb * P.slope;
    return P;
}

#if USE_ASYNC_LDS
// Toolchain types the async-copy builtin's pointer params as int4* in
// addrspace(1)/(3) (diagnostic from round 2), so cast to exactly that.
typedef int v4i __attribute__((ext_vector_type(4)));
typedef __attribute__((address_space(1))) v4i as1_v4i;
typedef __attribute__((address_space(3))) v4i as3_v4i;
#endif

// One thread per batch row; 32 threads (= 1 wave32) per block so each of the
// 4 waves lands on its own SIMD -> minimal per-wave issue contention for a
// purely latency-bound recurrence.
__global__ __launch_bounds__(32, 1)
void ssl_comp_scan(const float* __restrict__ x, float* __restrict__ out,
                   const float* __restrict__ p_thresh, const float* __restrict__ p_ratio,
                   const float* __restrict__ p_fb,
                   const float* __restrict__ p_uaf, const float* __restrict__ p_uas,
                   const float* __restrict__ p_usf, const float* __restrict__ p_uss,
                   int T, int B)
{
    const int b = blockIdx.x * blockDim.x + threadIdx.x;
    if (b >= B) return;

    const Params P = make_params(p_thresh, p_ratio, p_fb, p_uaf, p_uas, p_usf, p_uss);

    const float* xp = x   + (size_t)b * (size_t)T;
    float*       op = out + (size_t)b * (size_t)T;

    ScanState st{0.0f, 0.0f, 0.0f};
    int t = 0;

#if USE_ASYNC_LDS
    // ---- CDNA5 async global->LDS streaming pipeline (guaranteed prefetch) ----
    // Ring: S stages x K float4-units x 32 lanes. Per lane a stage = 32 samples
    // (128B = 1 cacheline). S-1 = 7 stages kept in flight via ASYNCcnt =>
    // ~224 samples (~5600 cycles of compute) of guaranteed read-ahead per wave.
    constexpr int CH = 32;          // samples per stage per lane
    constexpr int K  = CH / 4;      // float4 units per stage per lane (8)
    constexpr int S  = 8;           // ring depth (power of 2)
    __shared__ float4 ring[S * K * 32];   // 32 KB

    const int lane = threadIdx.x;
    const int nFull = T / CH;

    // issue one stage (K async b128 copies, one cacheline per lane)
    auto issue_stage = [&](int j) {
        const int slot = j & (S - 1);
        const float* g = xp + (size_t)j * CH;
#pragma unroll
        for (int k = 0; k < K; ++k) {
            __builtin_amdgcn_global_load_async_to_lds_b128(
                (as1_v4i*)(g + 4 * k),
                (as3_v4i*)&ring[(slot * K + k) * 32 + lane],
                /*offset=*/0, /*cpol=*/0);
        }
    };

    // consume one stage: 8 x (ds_load_b128 -> 4 recurrence steps -> b128 store)
    auto consume_stage = [&](int i) {
        const int slot = i & (S - 1);
        const int tb   = i * CH;
#pragma unroll
        for (int k = 0; k < K; ++k) {
            float4 c = ring[(slot * K + k) * 32 + lane];
            float4 o = step4(st, c, P);
            *(float4*)(op + tb + 4 * k) = o;
        }
    };

    // prologue: fill S-1 stages (56 async ops in flight, ASYNCcnt max is 63)
    const int pro = (S - 1 < nFull) ? (S - 1) : nFull;
    for (int j = 0; j < pro; ++j) issue_stage(j);

    // steady state: before consuming stage i we have issued i+S-1 stages, so
    // outstanding <= (S-2)*K guarantees stage i has fully landed in LDS.
    int i = 0;
    for (; i + (S - 1) < nFull; ++i) {
        __builtin_amdgcn_s_wait_asynccnt((S - 2) * K);   // = 48, compile-time imm
        consume_stage(i);
        issue_stage(i + S - 1);
    }
    // drain: constant-wait math no longer valid; flush everything once.
    if (nFull > 0) __builtin_amdgcn_s_wait_asynccnt(0);
    for (; i < nFull; ++i) consume_stage(i);

    t = nFull * CH;
#else
    // ---- fallback: register double-buffer + deep prefetch hint ----
    constexpr int U = 8;
    if (T >= 2 * U) {
        float4 c0 = ((const float4*)xp)[0];
        float4 c1 = ((const float4*)xp)[1];
        for (; t + 2 * U <= T; t += U) {
            float4 n0 = ((const float4*)(xp + t + U))[0];
            float4 n1 = ((const float4*)(xp + t + U))[1];
            int pf = t + 512;
            pf = pf < T ? pf : (T - 1);
            __builtin_prefetch(xp + pf, 0, 3);
            ((float4*)(op + t))[0] = step4(st, c0, P);
            ((float4*)(op + t))[1] = step4(st, c1, P);
            c0 = n0; c1 = n1;
        }
        ((float4*)(op + t))[0] = step4(st, c0, P);
        ((float4*)(op + t))[1] = step4(st, c1, P);
        t += U;
    }
#endif

    // scalar tail (empty for T = 131072)
    for (; t < T; ++t) {
        float s = (xp[t] - P.thresh) * P.slope;
        op[t] = step(st, s, P);
    }
}

} // anonymous namespace

extern "C" void kernel_launch(void* const* d_in, const int* in_sizes, int n_in,
                              void* d_out, int out_size, void* d_ws, size_t ws_size,
                              hipStream_t stream) {
    // Inputs (setup_inputs order):
    // 0: x_peak_dB (B*T f32), 1: comp_thresh, 2: ratio_logit, 3: fb_logit,
    // 4: u_T_af, 5: u_T_as, 6: u_T_sf, 7: u_T_ss  (all single-element f32)
    const float* x = (const float*)d_in[0];
    const int B = 128;                 // fixed by the reference harness
    const int n = in_sizes[0];
    const int T = n / B;

    dim3 block(32);                    // one wave32 per block
    dim3 grid((B + 31) / 32);          // 4 blocks -> 4 waves on 4 SIMDs/WGPs

    ssl_comp_scan<<<grid, block, 0, stream>>>(
        x, (float*)d_out,
        (const float*)d_in[1], (const float*)d_in[2], (const float*)d_in[3],
        (const float*)d_in[4], (const float*)d_in[5], (const float*)d_in[6],
        (const float*)d_in[7], T, B);
}